// RGCNLinkPredictor_58050777973059
// MI455X (gfx1250) — compile-verified
//
#include <hip/hip_runtime.h>

typedef __attribute__((ext_vector_type(2))) float v2f;
typedef __attribute__((ext_vector_type(8))) float v8f;

#define ND 50000
#define NP 50000
#define NE 800000
#define DIN 256
#define DHID 256
#define DOUT 128
#define KDIM 256  // all GEMMs have K = 256

__device__ __forceinline__ v8f wmma4(v2f a, v2f b, v8f c) {
  // V_WMMA_F32_16X16X4_F32: D = A(16x4) x B(4x16) + C(16x16), fp32
  return __builtin_amdgcn_wmma_f32_16x16x4_f32(false, a, false, b, (short)0, c,
                                               false, false);
}

// Stage one 256(K) x 128(N) weight panel into LDS, packed as float2 pairs of
// consecutive K rows: lds[kp*128 + n] = { W[2kp][tn0+n], W[2kp+1][tn0+n] }.
// All 256 threads participate (64 iterations, coalesced along n).
__device__ __forceinline__ void stage_panel(const float* __restrict__ W, int N,
                                            int tn0, float2* lds) {
  for (int i = threadIdx.x; i < 128 * 128; i += 256) {
    int kp = i >> 7;
    int n = i & 127;
    float2 v;
    v.x = W[(size_t)(2 * kp) * N + tn0 + n];
    v.y = W[(size_t)(2 * kp + 1) * N + tn0 + n];
    lds[i] = v;
  }
}

// out = act( [G1]  (A1/cnt1) @ W1
//          + [G2]   A2 @ W2
//          + [DIR]  Dadd/cntD            (direct add, weight pre-applied)
//          + bias )
// Block = 8 waves = 8 consecutive 16-row tiles of ONE 128-col strip; the
// weight panel for the strip lives in (dynamic) LDS, B frags are ds_load_b64.
template <bool G1, bool G2, bool DIR, bool RELU>
__global__ __launch_bounds__(256) void sage_gemm(
    const float* __restrict__ A1, const float* __restrict__ cnt1,
    const float* __restrict__ W1, const float* __restrict__ A2,
    const float* __restrict__ W2, const float* __restrict__ Dadd,
    const float* __restrict__ cntD, const float* __restrict__ bias,
    float* __restrict__ outp, int M, int N) {
  extern __shared__ float2 wpanel[];  // 128 kp x 128 n = 131072 bytes

  int nstrips = N >> 7;
  int strip = blockIdx.x % nstrips;
  int mblock = blockIdx.x / nstrips;
  int waveId = threadIdx.x >> 5;
  int task_m = mblock * 8 + waveId;
  int mtiles = M >> 4;
  bool active = task_m < mtiles;  // wave-uniform; do NOT return (barriers!)

  int lane = threadIdx.x & 31;
  int half = lane >> 4;  // K pair select (A/B), M half (C/D)
  int l16 = lane & 15;   // M row (A) / N col (B,C,D)
  int tm = task_m * 16;
  int tn0 = strip * 128;
  int am = tm + l16;

  v8f zero = {};
  v8f acc[8];
#pragma unroll
  for (int j = 0; j < 8; ++j) acc[j] = zero;

  if (G1) {
    stage_panel(W1, N, tn0, wpanel);
    __syncthreads();
    if (active) {
      float s1 = 1.0f / fmaxf(cnt1[am], 1.0f);
      const float* arow = A1 + (size_t)am * KDIM;
      for (int k = 0; k < KDIM; k += 4) {
        v2f a = *(const v2f*)(arow + k + 2 * half);
        a *= s1;  // fold 1/cnt mean into A rows
        const float2* wrow = wpanel + ((k >> 1) + half) * 128 + l16;
#pragma unroll
        for (int j = 0; j < 8; ++j) {
          float2 bb = wrow[j * 16];  // single ds_load_b64
          v2f b;
          b.x = bb.x;
          b.y = bb.y;
          acc[j] = wmma4(a, b, acc[j]);
        }
      }
    }
    if (G2) __syncthreads();  // panel re-staged below
  }

  if (G2) {
    stage_panel(W2, N, tn0, wpanel);
    __syncthreads();
    if (active) {
      const float* arow = A2 + (size_t)am * KDIM;
      for (int k = 0; k < KDIM; k += 4) {
        v2f a = *(const v2f*)(arow + k + 2 * half);
        const float2* wrow = wpanel + ((k >> 1) + half) * 128 + l16;
#pragma unroll
        for (int j = 0; j < 8; ++j) {
          float2 bb = wrow[j * 16];
          v2f b;
          b.x = bb.x;
          b.y = bb.y;
          acc[j] = wmma4(a, b, acc[j]);
        }
      }
    }
  }

  if (!active) return;

  float invD[8];
  if (DIR) {
#pragma unroll
    for (int r = 0; r < 8; ++r)
      invD[r] = 1.0f / fmaxf(cntD[tm + r + 8 * half], 1.0f);
  }

#pragma unroll
  for (int j = 0; j < 8; ++j) {
    int n = tn0 + j * 16 + l16;
    float bv = bias ? bias[n] : 0.0f;
#pragma unroll
    for (int r = 0; r < 8; ++r) {
      int m = tm + r + 8 * half;  // C/D layout: VGPR r -> M = r + 8*half
      float v = acc[j][r];
      if (DIR) v += Dadd[(size_t)m * N + n] * invD[r];
      v += bv;
      if (RELU) v = fmaxf(v, 0.0f);
      outp[(size_t)m * N + n] = v;
    }
  }
}

__global__ __launch_bounds__(256) void count_kernel(
    const int* __restrict__ dst, float* __restrict__ cnt, int n) {
  int i = blockIdx.x * blockDim.x + threadIdx.x;
  if (i < n) atomicAdd(&cnt[dst[i]], 1.0f);
}

// one wave per edge: agg[dst] += feat[src]  (d floats, coalesced)
__global__ __launch_bounds__(256) void scatter_add_kernel(
    const float* __restrict__ feat, const int* __restrict__ edge,
    float* __restrict__ agg, int n_edges, int d) {
  int e = (blockIdx.x * blockDim.x + threadIdx.x) >> 5;
  int lane = threadIdx.x & 31;
  if (e >= n_edges) return;
  int src = edge[e];
  int dst = edge[n_edges + e];
  const float* f = feat + (size_t)src * d;
  float* a = agg + (size_t)dst * d;
  for (int c = lane; c < d; c += 32) atomicAdd(&a[c], f[c]);
}

// one wave per 128-float row: x /= max(||x||, 1e-12)
__global__ __launch_bounds__(256) void l2norm_kernel(float* __restrict__ z,
                                                     int rows) {
  int r = (blockIdx.x * blockDim.x + threadIdx.x) >> 5;
  int lane = threadIdx.x & 31;
  if (r >= rows) return;
  float4* row = (float4*)(z + (size_t)r * DOUT);
  float4 v = row[lane];
  float s = v.x * v.x + v.y * v.y + v.z * v.z + v.w * v.w;
#pragma unroll
  for (int o = 16; o > 0; o >>= 1) s += __shfl_xor(s, o, 32);
  float inv = 1.0f / fmaxf(sqrtf(s), 1e-12f);
  v.x *= inv;
  v.y *= inv;
  v.z *= inv;
  v.w *= inv;
  row[lane] = v;
}

#define PANEL_BYTES (128 * 128 * sizeof(float2))  // 131072

static inline int gemm_grid(int M, int N) {
  int mblocks = (M / 16 + 7) / 8;
  return mblocks * (N / 128);
}

extern "C" void kernel_launch(void* const* d_in, const int* in_sizes, int n_in,
                              void* d_out, int out_size, void* d_ws,
                              size_t ws_size, hipStream_t stream) {
  const float* x_drug = (const float*)d_in[0];
  const float* x_prot = (const float*)d_in[1];
  const int* edge_dp = (const int*)d_in[2];
  const int* edge_pd = (const int*)d_in[3];
  const float* W1l_dp = (const float*)d_in[4];
  const float* W1r_dp = (const float*)d_in[5];
  const float* b1_dp = (const float*)d_in[6];
  const float* W1l_pd = (const float*)d_in[7];
  const float* W1r_pd = (const float*)d_in[8];
  const float* b1_pd = (const float*)d_in[9];
  const float* W2l_dp = (const float*)d_in[10];
  const float* W2r_dp = (const float*)d_in[11];
  const float* b2_dp = (const float*)d_in[12];
  const float* W2l_pd = (const float*)d_in[13];
  const float* W2r_pd = (const float*)d_in[14];
  const float* b2_pd = (const float*)d_in[15];

  float* z_drug = (float*)d_out;
  float* z_prot = z_drug + (size_t)ND * DOUT;

  // workspace layout (floats)
  float* ws = (float*)d_ws;
  float* cnt_prot = ws;                           // 50000
  float* cnt_drug = cnt_prot + NP;                // 50000
  float* agg_prot = cnt_drug + ND;                // 50000*256 (reused 50000*128)
  float* agg_drug = agg_prot + (size_t)NP * DIN;  // 50000*256 (reused 50000*128)
  float* h_prot = agg_drug + (size_t)ND * DIN;    // 50000*256
  float* h_drug = h_prot + (size_t)NP * DHID;     // 50000*256
  float* p_drug = h_drug + (size_t)ND * DHID;     // 50000*128
  float* p_prot = p_drug + (size_t)ND * DOUT;     // 50000*128

  // zero counts + layer-1 agg buffers (contiguous region)
  size_t zero1 = (size_t)(NP + ND) * sizeof(float) +
                 ((size_t)NP + ND) * DIN * sizeof(float);
  hipMemsetAsync(ws, 0, zero1, stream);

  // in-degree counts (dst = edge row 1)
  count_kernel<<<(NE + 255) / 256, 256, 0, stream>>>(edge_dp + NE, cnt_prot, NE);
  count_kernel<<<(NE + 255) / 256, 256, 0, stream>>>(edge_pd + NE, cnt_drug, NE);

  // layer-1 scatter: agg_prot += x_drug[src_dp]; agg_drug += x_prot[src_pd]
  scatter_add_kernel<<<NE / 8, 256, 0, stream>>>(x_drug, edge_dp, agg_prot, NE, DIN);
  scatter_add_kernel<<<NE / 8, 256, 0, stream>>>(x_prot, edge_pd, agg_drug, NE, DIN);

  // h = relu( (agg/cnt)@W1l + x@W1r + b1 )
  int g1 = gemm_grid(NP, DHID);
  sage_gemm<true, true, false, true><<<g1, 256, PANEL_BYTES, stream>>>(
      agg_prot, cnt_prot, W1l_dp, x_prot, W1r_dp, nullptr, nullptr, b1_dp,
      h_prot, NP, DHID);
  sage_gemm<true, true, false, true><<<g1, 256, PANEL_BYTES, stream>>>(
      agg_drug, cnt_drug, W1l_pd, x_drug, W1r_pd, nullptr, nullptr, b1_pd,
      h_drug, ND, DHID);

  // push layer-2 left GEMM before the scatter: p = h @ W2l (256 -> 128 cols)
  int g2 = gemm_grid(ND, DOUT);
  sage_gemm<false, true, false, false><<<g2, 256, PANEL_BYTES, stream>>>(
      nullptr, nullptr, nullptr, h_drug, W2l_dp, nullptr, nullptr, nullptr,
      p_drug, ND, DOUT);
  sage_gemm<false, true, false, false><<<g2, 256, PANEL_BYTES, stream>>>(
      nullptr, nullptr, nullptr, h_prot, W2l_pd, nullptr, nullptr, nullptr,
      p_prot, NP, DOUT);

  // reuse agg buffers as 128-wide layer-2 accumulators
  hipMemsetAsync(agg_prot, 0, (size_t)NP * DOUT * sizeof(float), stream);
  hipMemsetAsync(agg_drug, 0, (size_t)ND * DOUT * sizeof(float), stream);
  scatter_add_kernel<<<NE / 8, 256, 0, stream>>>(p_drug, edge_dp, agg_prot, NE, DOUT);
  scatter_add_kernel<<<NE / 8, 256, 0, stream>>>(p_prot, edge_pd, agg_drug, NE, DOUT);

  // z = agg2/cnt + h@W2r + b2   (weight already applied to agg2 term)
  sage_gemm<false, true, true, false><<<g2, 256, PANEL_BYTES, stream>>>(
      nullptr, nullptr, nullptr, h_prot, W2r_dp, agg_prot, cnt_prot, b2_dp,
      z_prot, NP, DOUT);
  sage_gemm<false, true, true, false><<<g2, 256, PANEL_BYTES, stream>>>(
      nullptr, nullptr, nullptr, h_drug, W2r_pd, agg_drug, cnt_drug, b2_pd,
      z_drug, ND, DOUT);

  // L2 normalize both outputs in place (contiguous 100000 x 128)
  int rows = ND + NP;
  l2norm_kernel<<<(rows * 32 + 255) / 256, 256, 0, stream>>>(z_drug, rows);
}